// HGTConv_64707977282168
// MI455X (gfx1250) — compile-verified
//
#include <hip/hip_runtime.h>
#include <math.h>

// HGTConv for MI455X (gfx1250): fp32 WMMA node GEMMs + wave-per-edge softmax/aggregation.

typedef __attribute__((ext_vector_type(2))) float v2f;
typedef __attribute__((ext_vector_type(8))) float v8f;

#define HID 128
#define DHEAD 32

// ---------------------------------------------------------------------------
// C[M x KO] = A[M x 128] @ B[128 x KO]  (+ optional bias[KO], optional C +=)
// KO = NT*16. One 256-thread block (8 waves) covers 128 rows; each wave does a
// 16-row stripe across all NT column tiles with V_WMMA_F32_16X16X4_F32.
// B is staged in LDS pre-paired as float2{B[2p][n],B[2p+1][n]} so each fragment
// is a single aligned ds_load_b64; pitch KO+16 float2 => the two half-waves hit
// disjoint 32-bank halves (conflict-free). A rows are clamped (not predicated):
// garbage rows only affect garbage output rows, which are never stored.
// ---------------------------------------------------------------------------
template <int NT>
__global__ __launch_bounds__(256) void gemm128_wmma(
    const float* __restrict__ A, const float* __restrict__ B,
    const float* __restrict__ bias, float* __restrict__ C,
    int M, int accumulate)
{
    constexpr int KO  = NT * 16;
    constexpr int LDP = KO + 16;         // float2 pitch: 2*LDP % 64 == 32 banks
    __shared__ v2f Bs[64 * LDP];

    const int tid = threadIdx.x;
    for (int i = tid; i < 64 * KO; i += 256) {
        const int p = i / KO;            // k-pair index (k = 2p, 2p+1)
        const int n = i & (KO - 1);
        v2f t;
        t.x = B[(2 * p) * KO + n];
        t.y = B[(2 * p + 1) * KO + n];
        Bs[p * LDP + n] = t;
    }
    __syncthreads();

    const int lane = tid & 31;
    const int wv   = tid >> 5;
    const int half = lane >> 4;          // 0: lanes 0-15, 1: lanes 16-31
    const int m16  = lane & 15;
    const int row0 = blockIdx.x * 128 + wv * 16;

    const int arow  = min(row0 + m16, M - 1);       // clamp, never predicate
    const float* ap = A + (size_t)arow * 128;

    v8f acc[NT];
#pragma unroll
    for (int t = 0; t < NT; ++t)
#pragma unroll
        for (int j = 0; j < 8; ++j) acc[t][j] = 0.0f;

#pragma unroll 4
    for (int ks = 0; ks < 32; ++ks) {
        const int k = ks * 4 + half * 2;            // lane holds A[m16][k..k+1]
        const v2f a = *reinterpret_cast<const v2f*>(ap + k);   // global_load_b64
        const v2f* brow = Bs + (ks * 2 + half) * LDP;
#pragma unroll
        for (int t = 0; t < NT; ++t) {
            const v2f b = brow[t * 16 + m16];       // ds_load_b64, aligned pair
            acc[t] = __builtin_amdgcn_wmma_f32_16x16x4_f32(
                false, a, false, b, (short)0, acc[t], false, false);
        }
    }

    // D layout: VGPR r, lane -> row (r + 8*half), col m16
#pragma unroll
    for (int t = 0; t < NT; ++t) {
        const float bval = bias ? bias[t * 16 + m16] : 0.0f;
#pragma unroll
        for (int r = 0; r < 8; ++r) {
            const int crow = row0 + half * 8 + r;
            if (crow < M) {
                const size_t idx = (size_t)crow * KO + t * 16 + m16;
                const float v = acc[t][r] + bval;
                C[idx] = accumulate ? (C[idx] + v) : v;
            }
        }
    }
}

// ------------------------- order-preserving float<->uint ---------------------
__device__ __forceinline__ unsigned enc_f(float f) {
    unsigned u = __float_as_uint(f);
    return (u & 0x80000000u) ? ~u : (u | 0x80000000u);
}
__device__ __forceinline__ float dec_f(unsigned k) {
    unsigned u = (k & 0x80000000u) ? (k ^ 0x80000000u) : ~k;
    return __uint_as_float(u);
}

// ---------------------------------------------------------------------------
// One wave per edge: per-head dot(q[col], rk[row]) with lane = channel,
// butterfly reduce, store scores[E,4], atomicMax of encoded score per (dst,h).
// ---------------------------------------------------------------------------
__global__ __launch_bounds__(256) void hgt_scores(
    const float* __restrict__ Q, const float* __restrict__ RK,
    const int* __restrict__ row, const int* __restrict__ col,
    int E, float* __restrict__ scores, unsigned* __restrict__ menc)
{
    const int lane = threadIdx.x & 31;
    const int e = blockIdx.x * (blockDim.x >> 5) + (threadIdx.x >> 5);
    if (e >= E) return;
    const int r = row[e], c = col[e];
    const float* qp = Q  + (size_t)c * HID;
    const float* kp = RK + (size_t)r * HID;

    float s[4];
#pragma unroll
    for (int h = 0; h < 4; ++h) {
        float p = qp[h * 32 + lane] * kp[h * 32 + lane];
#pragma unroll
        for (int o = 16; o > 0; o >>= 1) p += __shfl_xor(p, o, 32);
        s[h] = p * 0.17677669529663687f;     // 1/sqrt(32)
    }
    if (lane == 0) {
        float4 sv = make_float4(s[0], s[1], s[2], s[3]);
        *reinterpret_cast<float4*>(scores + (size_t)e * 4) = sv;
#pragma unroll
        for (int h = 0; h < 4; ++h)
            atomicMax(menc + (size_t)c * 4 + h, enc_f(s[h]));
    }
}

// ---------------------------------------------------------------------------
// One wave per edge: e = exp(s - max); num[col] += rv[row]*e ; den[col] += e.
// Softmax denominator is per-(dst,head) constant -> normalize later per node.
// ---------------------------------------------------------------------------
__global__ __launch_bounds__(256) void hgt_accum(
    const float* __restrict__ RV,
    const int* __restrict__ row, const int* __restrict__ col,
    const float* __restrict__ scores, const unsigned* __restrict__ menc,
    int E, float* __restrict__ num, float* __restrict__ den)
{
    const int lane = threadIdx.x & 31;
    const int e = blockIdx.x * (blockDim.x >> 5) + (threadIdx.x >> 5);
    if (e >= E) return;
    const int r = row[e], c = col[e];

    float w[4];
#pragma unroll
    for (int h = 0; h < 4; ++h) {
        const float s = scores[(size_t)e * 4 + h];
        const float m = dec_f(menc[(size_t)c * 4 + h]);
        w[h] = expf(s - m);
    }
    const float* vp = RV + (size_t)r * HID;
#pragma unroll
    for (int h = 0; h < 4; ++h)
        unsafeAtomicAdd(num + (size_t)c * HID + h * 32 + lane,
                        vp[h * 32 + lane] * w[h]);
    if (lane < 4)
        unsafeAtomicAdd(den + (size_t)c * 4 + lane, w[lane]);
}

// out_p[n,d] = num_w/den_w + num_c/den_c   (0 for empty segments)
__global__ __launch_bounds__(256) void hgt_finalize(
    const float* __restrict__ num_w, const float* __restrict__ den_w,
    const float* __restrict__ num_c, const float* __restrict__ den_c,
    float* __restrict__ out_p, int Nn)
{
    const size_t i = (size_t)blockIdx.x * blockDim.x + threadIdx.x;
    if (i >= (size_t)Nn * HID) return;
    const int n = (int)(i >> 7);
    const int h = (int)((i & 127) >> 5);
    const float dw = den_w[(size_t)n * 4 + h];
    const float dc = den_c[(size_t)n * 4 + h];
    const float aw = dw > 0.0f ? num_w[i] / dw : 0.0f;
    const float ac = dc > 0.0f ? num_c[i] / dc : 0.0f;
    out_p[i] = aw + ac;
}

__global__ void zero_f32(float* __restrict__ p, long n)
{
    for (long i = (long)blockIdx.x * blockDim.x + threadIdx.x; i < n;
         i += (long)gridDim.x * blockDim.x)
        p[i] = 0.0f;
}

// ---------------------------------------------------------------------------
extern "C" void kernel_launch(void* const* d_in, const int* in_sizes, int n_in,
                              void* d_out, int out_size, void* d_ws, size_t ws_size,
                              hipStream_t stream)
{
    const float* x_a   = (const float*)d_in[0];
    const float* x_p   = (const float*)d_in[1];
    const int*   ew    = (const int*)d_in[2];
    const int*   ec    = (const int*)d_in[3];
    const float* wk_a  = (const float*)d_in[5];
    const float* wv_a  = (const float*)d_in[6];
    const float* wq_p  = (const float*)d_in[7];
    const float* wk_p  = (const float*)d_in[8];
    const float* wv_p  = (const float*)d_in[9];
    const float* wrk_w = (const float*)d_in[10];
    const float* wrv_w = (const float*)d_in[11];
    const float* wrk_c = (const float*)d_in[12];
    const float* wrv_c = (const float*)d_in[13];
    const float* bo_a  = (const float*)d_in[15];
    const float* wo_p  = (const float*)d_in[16];
    const float* bo_p  = (const float*)d_in[17];
    const float* wr_a  = (const float*)d_in[18];
    const float* wr_p  = (const float*)d_in[19];

    const int Nn  = in_sizes[0] / 128;      // 50000
    const int E_w = in_sizes[2] / 2;        // 600000
    const int E_c = in_sizes[3] / 2;

    // ---- workspace layout (floats) ----
    float* ws = (float*)d_ws;
    size_t o = 0;
    float* Q    = ws + o; o += (size_t)Nn * HID;
    float* RKW  = ws + o; o += (size_t)Nn * HID;
    float* RVW  = ws + o; o += (size_t)Nn * HID;
    float* RKC  = ws + o; o += (size_t)Nn * HID;
    float* RVC  = ws + o; o += (size_t)Nn * HID;
    float* OUTP = ws + o; o += (size_t)Nn * HID;
    float* W1   = ws + o; o += 128 * 128;   // wk_a @ wrk_w
    float* W2   = ws + o; o += 128 * 128;   // wv_a @ wrv_w
    float* W3   = ws + o; o += 128 * 128;   // wk_p @ wrk_c
    float* W4   = ws + o; o += 128 * 128;   // wv_p @ wrv_c
    float* SCW  = ws + o; o += (size_t)E_w * 4;
    float* SCC  = ws + o; o += (size_t)E_c * 4;
    float*    NUMW = ws + o; o += (size_t)Nn * HID;
    float*    NUMC = ws + o; o += (size_t)Nn * HID;
    float*    DENW = ws + o; o += (size_t)Nn * 4;
    float*    DENC = ws + o; o += (size_t)Nn * 4;
    unsigned* MENW = (unsigned*)(ws + o); o += (size_t)Nn * 4;
    unsigned* MENC = (unsigned*)(ws + o); o += (size_t)Nn * 4;
    const long zcount = (long)Nn * (HID + HID + 4 + 4 + 4 + 4);  // NUM..MENC contiguous

    float* fin_a = (float*)d_out;                       // [N, 32]
    float* fin_p = (float*)d_out + (size_t)Nn * DHEAD;  // [N, 32]

    const int gridM = (Nn + 127) / 128;

    // 1) zero accumulators / encoded maxima (0 < enc(x) for every finite x)
    zero_f32<<<4096, 256, 0, stream>>>(NUMW, zcount);

    // 2) fuse per-relation weights: W = wk @ wrk (128x128x128 each)
    gemm128_wmma<8><<<1, 256, 0, stream>>>(wk_a, wrk_w, nullptr, W1, 128, 0);
    gemm128_wmma<8><<<1, 256, 0, stream>>>(wv_a, wrv_w, nullptr, W2, 128, 0);
    gemm128_wmma<8><<<1, 256, 0, stream>>>(wk_p, wrk_c, nullptr, W3, 128, 0);
    gemm128_wmma<8><<<1, 256, 0, stream>>>(wv_p, wrv_c, nullptr, W4, 128, 0);

    // 3) node projections (fused): Q = x_p@wq_p, RK*/RV* = x@(wk@wrk) etc.
    gemm128_wmma<8><<<gridM, 256, 0, stream>>>(x_p, wq_p, nullptr, Q,   Nn, 0);
    gemm128_wmma<8><<<gridM, 256, 0, stream>>>(x_a, W1,   nullptr, RKW, Nn, 0);
    gemm128_wmma<8><<<gridM, 256, 0, stream>>>(x_a, W2,   nullptr, RVW, Nn, 0);
    gemm128_wmma<8><<<gridM, 256, 0, stream>>>(x_p, W3,   nullptr, RKC, Nn, 0);
    gemm128_wmma<8><<<gridM, 256, 0, stream>>>(x_p, W4,   nullptr, RVC, Nn, 0);

    // 4) edge scores + segment max (writes: author->paper, cites: paper->paper)
    const int gridEw = (E_w + 7) / 8, gridEc = (E_c + 7) / 8;
    hgt_scores<<<gridEw, 256, 0, stream>>>(Q, RKW, ew, ew + E_w, E_w, SCW, MENW);
    hgt_scores<<<gridEc, 256, 0, stream>>>(Q, RKC, ec, ec + E_c, E_c, SCC, MENC);

    // 5) exp + unnormalized scatter-accumulate
    hgt_accum<<<gridEw, 256, 0, stream>>>(RVW, ew, ew + E_w, SCW, MENW, E_w, NUMW, DENW);
    hgt_accum<<<gridEc, 256, 0, stream>>>(RVC, ec, ec + E_c, SCC, MENC, E_c, NUMC, DENC);

    // 6) normalize + combine relations
    hgt_finalize<<<(int)(((size_t)Nn * HID + 255) / 256), 256, 0, stream>>>(
        NUMW, DENW, NUMC, DENC, OUTP, Nn);

    // 7) outputs:
    //    fin_p = out_p@wo_p + bo_p + x_p@wr_p ; fin_a = x_a@wr_a + bo_a (out_a == 0)
    gemm128_wmma<2><<<gridM, 256, 0, stream>>>(OUTP, wo_p, bo_p, fin_p, Nn, 0);
    gemm128_wmma<2><<<gridM, 256, 0, stream>>>(x_p,  wr_p, nullptr, fin_p, Nn, 1);
    gemm128_wmma<2><<<gridM, 256, 0, stream>>>(x_a,  wr_a, bo_a, fin_a, Nn, 0);
}